// DiagonalVariational_12945031431044
// MI455X (gfx1250) — compile-verified
//
#include <hip/hip_runtime.h>
#include <hip/hip_bf16.h>

// z[n,i] = m[i] + std_normal[n,i] * (diag_L[i] + JITTER)
// Pure HBM-bandwidth-bound streaming FMA: ~512 MB moved, ~0.13 GFLOP.
// Strategy: b128 vector path, NT cache hints on the 256MB single-use streams,
// temporal (L2-resident) loads for the 64KB m/diag_L vectors, gfx1250
// global_prefetch on the streaming operand.

#ifndef JITTER_F
#define JITTER_F 1e-6f
#endif

// Native clang vector (required by __builtin_nontemporal_load/store).
typedef float v4f __attribute__((ext_vector_type(4)));

constexpr int TPB            = 256; // 8 wave32 per block
constexpr int ROWS_PER_BLOCK = 8;

__global__ __launch_bounds__(TPB) void diag_variational_kernel(
    const float* __restrict__ m,
    const float* __restrict__ diag_L,
    const float* __restrict__ std_normal,
    float* __restrict__ out,
    int d4,      // D / 4  (float4 columns per row)
    int n_rows)  // N_SAMPLE
{
    const int c4   = blockIdx.x * TPB + threadIdx.x;  // float4 column index
    const int row0 = blockIdx.y * ROWS_PER_BLOCK;
    if (c4 >= d4) return;

    // Small per-column vectors: regular temporal loads (64KB each, reused
    // n_rows times -> resident in L2 / WGP$).
    const v4f m4 = reinterpret_cast<const v4f*>(m)[c4];
    v4f s4       = reinterpret_cast<const v4f*>(diag_L)[c4];
    s4 += (v4f)JITTER_F;

    const v4f* __restrict__ src =
        reinterpret_cast<const v4f*>(std_normal) + (size_t)row0 * d4 + c4;
    v4f* __restrict__ dst =
        reinterpret_cast<v4f*>(out) + (size_t)row0 * d4 + c4;

#pragma unroll
    for (int r = 0; r < ROWS_PER_BLOCK; ++r) {
        const int row = row0 + r;
        if (row >= n_rows) return;

        // Prefetch next row's element for this lane (speculative: OOB at the
        // tail is silently dropped per ISA prefetch semantics).
        __builtin_prefetch(src + d4, 0, 0);

        // Single-use 256MB stream: non-temporal load (don't pollute caches).
        v4f x = __builtin_nontemporal_load(src);

        v4f z;
        z.x = fmaf(x.x, s4.x, m4.x);
        z.y = fmaf(x.y, s4.y, m4.y);
        z.z = fmaf(x.z, s4.z, m4.z);
        z.w = fmaf(x.w, s4.w, m4.w);

        // Single-use 256MB output stream: non-temporal store.
        __builtin_nontemporal_store(z, dst);

        src += d4;
        dst += d4;
    }
}

extern "C" void kernel_launch(void* const* d_in, const int* in_sizes, int n_in,
                              void* d_out, int out_size, void* d_ws, size_t ws_size,
                              hipStream_t stream) {
    const float* m          = (const float*)d_in[0];
    const float* diag_L     = (const float*)d_in[1];
    const float* std_normal = (const float*)d_in[2];
    float* out              = (float*)d_out;

    const int D  = in_sizes[0];          // 16384
    const int N  = in_sizes[2] / D;      // 4096
    const int d4 = D / 4;                // 4096 float4 columns

    dim3 block(TPB, 1, 1);
    dim3 grid((d4 + TPB - 1) / TPB,
              (N + ROWS_PER_BLOCK - 1) / ROWS_PER_BLOCK, 1);

    diag_variational_kernel<<<grid, block, 0, stream>>>(
        m, diag_L, std_normal, out, d4, N);
}